// MessagePassing_41240275976751
// MI455X (gfx1250) — compile-verified
//
#include <hip/hip_runtime.h>
#include <stdint.h>

// MessagePassing: out = segment_sum(x[edge_index[1]], edge_index[0], 50000)
// Memory/atomic-bound: x (12.8MB) and out (12.8MB) are L2-resident on MI455X
// (192MB L2). Strategy: b128 gathers + hardware f32 atomics; edge-index tiles
// staged into LDS via CDNA5 async global->LDS copies, double-buffered.

#define THREADS 256
#define TILE_E  64   // edges per staged tile
#define LPE     16   // lanes per edge: 64 feats / 4 per float4

__device__ __forceinline__ uint32_t lds_addr_of(const void* p) {
  // Low 32 bits of a flat shared-aperture address are the LDS byte address.
  return (uint32_t)(uintptr_t)p;
}

__device__ __forceinline__ void async_copy_b128(uint32_t lds_byte_addr,
                                                const void* gaddr) {
  asm volatile("global_load_async_to_lds_b128 %0, %1, off"
               :: "v"(lds_byte_addr), "v"((uint64_t)(uintptr_t)gaddr)
               : "memory");
}

__device__ __forceinline__ void wait_async0() {
#if __has_builtin(__builtin_amdgcn_s_wait_asynccnt)
  __builtin_amdgcn_s_wait_asynccnt(0);
#else
  asm volatile("s_wait_asynccnt 0x0" ::: "memory");
#endif
}

__global__ __launch_bounds__(THREADS)
void MP_scatter_add_kernel(const float* __restrict__ x,
                           const int* __restrict__ eidx,   // [2, E] flat
                           float* __restrict__ out,
                           int E, int numTiles) {
  // [parity][ 0..63 = dst ids, 64..127 = src ids ]
  __shared__ __align__(16) int sIdx[2][2 * TILE_E];

  const int tid = threadIdx.x;
  const int* dstRow = eidx;
  const int* srcRow = eidx + E;

  int t = blockIdx.x;
  if (t >= numTiles) return;

  auto stage = [&](int tile, int par) {
    if (tid < 32) {
      int base = tile * TILE_E;
      if (base + TILE_E <= E) {
        // 32 lanes x 16B async copies: lanes 0..15 dst row, 16..31 src row
        int lane = tid & 15;
        const int* g = (tid < 16) ? (dstRow + base + 4 * lane)
                                  : (srcRow + base + 4 * lane);
        int ldsOff = (tid < 16) ? (4 * lane) : (TILE_E + 4 * lane);
        async_copy_b128(lds_addr_of(&sIdx[par][ldsOff]), g);
      } else {
        // tail tile: scalar stage (bounds-checked)
        for (int j = tid; j < TILE_E; j += 32) {
          int e = base + j;
          if (e < E) {
            sIdx[par][j] = dstRow[e];
            sIdx[par][TILE_E + j] = srcRow[e];
          }
        }
      }
    }
  };

  // Prologue: stage first tile into buffer 0.
  stage(t, 0);
  wait_async0();
  __syncthreads();

  int par = 0;
  const int stride = gridDim.x;
  for (; t < numTiles; t += stride) {
    // Kick off next tile's async staging into the other buffer.
    int tn = t + stride;
    if (tn < numTiles) stage(tn, par ^ 1);

    // Consume current tile: 64 edges x 16 chunk-tasks = 1024 tasks / 256 thr.
    int base = t * TILE_E;
    if (base + TILE_E <= E) {
      // Full tile (always taken for E % TILE_E == 0): no per-lane bounds
      // checks, so the load + 4-atomic sequence issues as a tight clause.
#pragma unroll
      for (int i = 0; i < (TILE_E * LPE) / THREADS; ++i) {
        int task = tid + i * THREADS;
        int e = task >> 4;     // local edge 0..63
        int c = task & 15;     // float4 chunk 0..15
        int dst = sIdx[par][e];
        int src = sIdx[par][TILE_E + e];
        const float4 v = reinterpret_cast<const float4*>(x)[src * 16 + c];
        float* o = out + (size_t)dst * 64 + (size_t)c * 4;
        atomicAdd(o + 0, v.x);
        atomicAdd(o + 1, v.y);
        atomicAdd(o + 2, v.z);
        atomicAdd(o + 3, v.w);
      }
    } else {
#pragma unroll
      for (int i = 0; i < (TILE_E * LPE) / THREADS; ++i) {
        int task = tid + i * THREADS;
        int e = task >> 4;
        int c = task & 15;
        if (base + e < E) {
          int dst = sIdx[par][e];
          int src = sIdx[par][TILE_E + e];
          const float4 v = reinterpret_cast<const float4*>(x)[src * 16 + c];
          float* o = out + (size_t)dst * 64 + (size_t)c * 4;
          atomicAdd(o + 0, v.x);
          atomicAdd(o + 1, v.y);
          atomicAdd(o + 2, v.z);
          atomicAdd(o + 3, v.w);
        }
      }
    }

    // Ensure next tile's async writes landed, and all lanes done reading
    // this buffer before it is re-staged in the next iteration.
    wait_async0();
    __syncthreads();
    par ^= 1;
  }
}

__global__ void MP_zero_kernel(float4* __restrict__ o, int n4) {
  int i = blockIdx.x * blockDim.x + threadIdx.x;
  if (i < n4) o[i] = make_float4(0.f, 0.f, 0.f, 0.f);
}

extern "C" void kernel_launch(void* const* d_in, const int* in_sizes, int n_in,
                              void* d_out, int out_size, void* d_ws, size_t ws_size,
                              hipStream_t stream) {
  const float* x    = (const float*)d_in[0];
  const int*   eidx = (const int*)d_in[1];    // [2, E] row-major
  float*       out  = (float*)d_out;

  const int E  = in_sizes[1] / 2;
  const int n4 = out_size / 4;                // out_size = 50000*64, /4 exact

  // d_out is poisoned; atomics need zeros.
  MP_zero_kernel<<<(n4 + THREADS - 1) / THREADS, THREADS, 0, stream>>>(
      (float4*)d_out, n4);

  const int numTiles = (E + TILE_E - 1) / TILE_E;
  int blocks = numTiles < 2048 ? numTiles : 2048;
  MP_scatter_add_kernel<<<blocks, THREADS, 0, stream>>>(x, eidx, out, E, numTiles);
}